// SelfAttention_46583215292625
// MI455X (gfx1250) — compile-verified
//
#include <hip/hip_runtime.h>
#include <math.h>

typedef __attribute__((ext_vector_type(2))) float v2f;
typedef __attribute__((ext_vector_type(8))) float v8f;

#define WMMA_F32(a, b, c) \
  __builtin_amdgcn_wmma_f32_16x16x4_f32(false, (a), false, (b), (short)0, (c), false, false)

#define NEG_INF (-__builtin_inff())

#define B_SZ 2
#define S_LEN 2048
#define D_MODEL 1024
#define H_CNT 16
#define DH 64

// ---------------------------------------------------------------------------
// CDNA5 async global->LDS DMA (ASYNCcnt-tracked). The generic pointer's low
// 32 bits are the LDS byte offset (shared aperture has zero low bits); the
// global address is the full 64-bit flat address. 16-byte alignment required.
// ---------------------------------------------------------------------------
__device__ __forceinline__ void async_copy_b128(void* lds, const void* gaddr) {
  unsigned lds_off = (unsigned)(unsigned long long)lds;
  asm volatile("global_load_async_to_lds_b128 %0, %1, off"
               :: "v"(lds_off), "v"(gaddr)
               : "memory");
}
__device__ __forceinline__ void wait_async0() {
  asm volatile("s_wait_asynccnt 0x0" ::: "memory");
}

// ---------------------------------------------------------------------------
// Generic C[M,N] = A[M,K] @ B[K,N] + bias[N], f32, WMMA 16x16x4.
// Block: 256 threads (8 waves). Tile: 64 rows x 128 cols. K staged 16 at a time
// via async-to-LDS DMA. Wave w: rows (w&1)*32, cols (w>>1)*32 -> 2x2 fragments.
// ---------------------------------------------------------------------------
__global__ __launch_bounds__(256)
void gemm_bias_wmma(const float* __restrict__ A, const float* __restrict__ B,
                    const float* __restrict__ bias, float* __restrict__ C,
                    int M, int N, int K) {
  __shared__ float As[64][20];    // stride 80B: 16B-aligned + conflict-free columns
  __shared__ float Bs[16][132];   // stride 528B: 16B-aligned, halves on distinct banks

  const int tid  = threadIdx.x;
  const int lane = tid & 31;
  const int wave = tid >> 5;      // 0..7
  const int wr   = wave & 1;      // 32-row group
  const int wc   = wave >> 1;     // 32-col group
  const int m0   = blockIdx.y * 64;
  const int n0   = blockIdx.x * 128;

  const int hi = lane >> 4;       // lane half
  const int ln = lane & 15;
  const int kb = hi * 2;          // K pair base within a 4-wide K step

  // per-thread staging coordinates
  const int ar = tid >> 2;                 // A row 0..63
  const int ac = (tid & 3) * 4;            // A col 0,4,8,12
  const int br = tid >> 5;                 // B row 0..7 (and +8)
  const int bc = (tid & 31) * 4;           // B col 0..124

  v8f acc00 = {}, acc01 = {}, acc10 = {}, acc11 = {};

  for (int k0 = 0; k0 < K; k0 += 16) {
    // --- async DMA: A tile 64x16, B tile 16x128 straight into LDS ---
    async_copy_b128(&As[ar][ac], A + (size_t)(m0 + ar) * K + k0 + ac);
    async_copy_b128(&Bs[br][bc],     B + (size_t)(k0 + br) * N + n0 + bc);
    async_copy_b128(&Bs[br + 8][bc], B + (size_t)(k0 + br + 8) * N + n0 + bc);
    wait_async0();
    __syncthreads();

    #pragma unroll
    for (int kk = 0; kk < 16; kk += 4) {
      // A fragments (16x4): lane holds A[m=ln][kk+kb], A[m=ln][kk+kb+1]
      v2f a0, a1, b0, b1;
      a0.x = As[wr * 32 + ln][kk + kb];
      a0.y = As[wr * 32 + ln][kk + kb + 1];
      a1.x = As[wr * 32 + 16 + ln][kk + kb];
      a1.y = As[wr * 32 + 16 + ln][kk + kb + 1];
      // B fragments (4x16): lane holds B[kk+kb][n=ln], B[kk+kb+1][n=ln]
      b0.x = Bs[kk + kb][wc * 32 + ln];
      b0.y = Bs[kk + kb + 1][wc * 32 + ln];
      b1.x = Bs[kk + kb][wc * 32 + 16 + ln];
      b1.y = Bs[kk + kb + 1][wc * 32 + 16 + ln];

      acc00 = WMMA_F32(a0, b0, acc00);
      acc01 = WMMA_F32(a0, b1, acc01);
      acc10 = WMMA_F32(a1, b0, acc10);
      acc11 = WMMA_F32(a1, b1, acc11);
    }
    __syncthreads();
  }

  // --- store: C/D layout: VGPR r -> row r + hi*8, col = ln ---
  v8f* accs[4] = {&acc00, &acc01, &acc10, &acc11};
  #pragma unroll
  for (int i = 0; i < 2; ++i) {
    #pragma unroll
    for (int j = 0; j < 2; ++j) {
      v8f a = *accs[i * 2 + j];
      int rowb = m0 + wr * 32 + i * 16 + hi * 8;
      int colb = n0 + wc * 32 + j * 16 + ln;
      float bv = bias[colb];
      #pragma unroll
      for (int r = 0; r < 8; ++r)
        C[(size_t)(rowb + r) * N + colb] = a[r] + bv;
    }
  }
}

// ---------------------------------------------------------------------------
// Causal flash attention over qkv [B,S,3D] -> attn [B,S,D], f32 WMMA.
// Block: 128 threads (4 waves) = one (b,h) and 64 query rows; wave w owns
// query rows [q0+16w, q0+16w+16). Keys streamed 16 at a time (async DMA into
// LDS) with online softmax; Q pre-scaled by 1/sqrt(dh).
// ---------------------------------------------------------------------------
__global__ __launch_bounds__(128)
void flash_attn_wmma(const float* __restrict__ qkv, float* __restrict__ attn) {
  __shared__ float Qs[64][65];
  __shared__ float Ks[16][68];   // stride 272B: 16B-aligned, conflict-free
  __shared__ float Vs[16][68];
  __shared__ float Ps[4][16][17];

  const int tid  = threadIdx.x;
  const int lane = tid & 31;
  const int wave = tid >> 5;        // 0..3
  const int hi   = lane >> 4;
  const int ln   = lane & 15;
  const int kb   = hi * 2;

  const int qtile = blockIdx.x;     // 0..S/64-1
  const int bh    = blockIdx.y;     // 0..B*H-1
  const int b     = bh / H_CNT;
  const int h     = bh % H_CNT;
  const int q0    = qtile * 64;

  const size_t rstr = 3 * D_MODEL;
  const float* qbase = qkv + (size_t)b * S_LEN * rstr + h * DH;
  const float* kbase = qbase + D_MODEL;
  const float* vbase = qbase + 2 * D_MODEL;

  // stage Q tile, pre-scaled by 1/sqrt(64) (scalar path: needs the multiply)
  {
    int c = (tid & 15) * 4;
    for (int r = tid >> 4; r < 64; r += 8) {
      const float* src = qbase + (size_t)(q0 + r) * rstr + c;
      Qs[r][c + 0] = src[0] * 0.125f;
      Qs[r][c + 1] = src[1] * 0.125f;
      Qs[r][c + 2] = src[2] * 0.125f;
      Qs[r][c + 3] = src[3] * 0.125f;
    }
  }

  v8f o0 = {}, o1 = {}, o2 = {}, o3 = {};
  float mrow[8], lrow[8];
  #pragma unroll
  for (int r = 0; r < 8; ++r) { mrow[r] = NEG_INF; lrow[r] = 0.0f; }

  const int ntiles = (qtile + 1) * 4;   // causal key-tile bound for this block
  const int qw_last = q0 + wave * 16 + 15;

  const int sr = tid >> 4;              // staging row 0..7 (and +8)
  const int scc = (tid & 15) * 4;       // staging col

  for (int j = 0; j < ntiles; ++j) {
    // --- async DMA: K,V tiles 16x64 straight into LDS ---
    async_copy_b128(&Ks[sr][scc],     kbase + (size_t)(j * 16 + sr) * rstr + scc);
    async_copy_b128(&Ks[sr + 8][scc], kbase + (size_t)(j * 16 + sr + 8) * rstr + scc);
    async_copy_b128(&Vs[sr][scc],     vbase + (size_t)(j * 16 + sr) * rstr + scc);
    async_copy_b128(&Vs[sr + 8][scc], vbase + (size_t)(j * 16 + sr + 8) * rstr + scc);
    wait_async0();
    __syncthreads();

    const bool active = (j * 16) <= qw_last;   // wave-uniform causal skip

    if (active) {
      // scores (16q x 16k) = Q . K^T, K dim = dh = 64
      v8f sc = {};
      const int qr = wave * 16;
      #pragma unroll
      for (int d0 = 0; d0 < DH; d0 += 4) {
        v2f a, bf;
        a.x  = Qs[qr + ln][d0 + kb];
        a.y  = Qs[qr + ln][d0 + kb + 1];
        bf.x = Ks[ln][d0 + kb];          // B[k_dim=d][n=key] = K[key][d]
        bf.y = Ks[ln][d0 + kb + 1];
        sc = WMMA_F32(a, bf, sc);
      }

      // causal mask + online softmax (row = q0 + 16*wave + hi*8 + r, col = 16j + ln)
      const int qab = q0 + wave * 16 + hi * 8;
      const int key = j * 16 + ln;
      #pragma unroll
      for (int r = 0; r < 8; ++r) {
        float s = sc[r];
        if (key > qab + r) s = NEG_INF;
        float v = s;
        #pragma unroll
        for (int off = 1; off < 16; off <<= 1) v = fmaxf(v, __shfl_xor(v, off, 32));
        float mnew  = fmaxf(mrow[r], v);
        float alpha = __expf(mrow[r] - mnew);
        float p     = __expf(s - mnew);
        float ps    = p;
        #pragma unroll
        for (int off = 1; off < 16; off <<= 1) ps += __shfl_xor(ps, off, 32);
        lrow[r] = lrow[r] * alpha + ps;
        mrow[r] = mnew;
        o0[r] *= alpha; o1[r] *= alpha; o2[r] *= alpha; o3[r] *= alpha;
        Ps[wave][hi * 8 + r][ln] = p;    // C-layout -> LDS (A-layout source)
      }
    }
    __syncthreads();   // P visible; also orders vs. next K/V overwrite path

    if (active) {
      // O += P (16x16) . V (16x64), 4 column tiles of 16
      #pragma unroll
      for (int ks = 0; ks < 16; ks += 4) {
        v2f a;
        a.x = Ps[wave][ln][ks + kb];
        a.y = Ps[wave][ln][ks + kb + 1];
        v2f bf;
        bf.x = Vs[ks + kb][0 * 16 + ln]; bf.y = Vs[ks + kb + 1][0 * 16 + ln];
        o0 = WMMA_F32(a, bf, o0);
        bf.x = Vs[ks + kb][1 * 16 + ln]; bf.y = Vs[ks + kb + 1][1 * 16 + ln];
        o1 = WMMA_F32(a, bf, o1);
        bf.x = Vs[ks + kb][2 * 16 + ln]; bf.y = Vs[ks + kb + 1][2 * 16 + ln];
        o2 = WMMA_F32(a, bf, o2);
        bf.x = Vs[ks + kb][3 * 16 + ln]; bf.y = Vs[ks + kb + 1][3 * 16 + ln];
        o3 = WMMA_F32(a, bf, o3);
      }
    }
    __syncthreads();
  }

  // write normalized output: attn[b, q, h*64 + d]
  const v8f* os[4] = {&o0, &o1, &o2, &o3};
  #pragma unroll
  for (int ct = 0; ct < 4; ++ct) {
    v8f a = *os[ct];
    #pragma unroll
    for (int r = 0; r < 8; ++r) {
      int q = q0 + wave * 16 + hi * 8 + r;
      int d = h * DH + ct * 16 + ln;
      attn[((size_t)b * S_LEN + q) * D_MODEL + d] = a[r] / lrow[r];
    }
  }
}

// ---------------------------------------------------------------------------
extern "C" void kernel_launch(void* const* d_in, const int* in_sizes, int n_in,
                              void* d_out, int out_size, void* d_ws, size_t ws_size,
                              hipStream_t stream) {
  const float* x     = (const float*)d_in[0];   // [2,2048,1024]
  const float* w_in  = (const float*)d_in[1];   // [1024,3072]
  const float* b_in  = (const float*)d_in[2];   // [3072]
  const float* w_out = (const float*)d_in[3];   // [1024,1024]
  const float* b_out = (const float*)d_in[4];   // [1024]
  float* out = (float*)d_out;                   // [2,2048,1024]

  const int M = B_SZ * S_LEN;                   // 4096
  float* qkv  = (float*)d_ws;                   // M * 3072 floats
  float* attn = qkv + (size_t)M * 3 * D_MODEL;  // M * 1024 floats

  // 1) qkv = x @ w_in + b_in
  dim3 g1(3 * D_MODEL / 128, M / 64);
  gemm_bias_wmma<<<g1, 256, 0, stream>>>(x, w_in, b_in, qkv, M, 3 * D_MODEL, D_MODEL);

  // 2) causal flash attention
  dim3 g2(S_LEN / 64, B_SZ * H_CNT);
  flash_attn_wmma<<<g2, 128, 0, stream>>>(qkv, attn);

  // 3) out = attn @ w_out + b_out
  dim3 g3(D_MODEL / 128, M / 64);
  gemm_bias_wmma<<<g3, 256, 0, stream>>>(attn, w_out, b_out, out, M, D_MODEL, D_MODEL);
}